// VNN_RI_Feature_82360292868837
// MI455X (gfx1250) — compile-verified
//
#include <hip/hip_runtime.h>
#include <hip/hip_bf16.h>
#include <math.h>

#define NB 4
#define NN 4096
#define NC 64
#define NK 20
#define COLSB (3 * NN)               // 12288 columns per batch in (B,C,3,N)
#define TSZ ((size_t)NB * NC * 3 * NN)
#define EPSF 1e-6f

typedef float v2f __attribute__((ext_vector_type(2)));
typedef float v8f __attribute__((ext_vector_type(8)));

// ---------------------------------------------------------------------------
// center[b][3] = mean over N of x[b,:,3]
// ---------------------------------------------------------------------------
__global__ __launch_bounds__(256) void center_kernel(const float* __restrict__ x,
                                                     float* __restrict__ center) {
  __shared__ float sm[256][3];
  int b = blockIdx.x;
  float a0 = 0.f, a1 = 0.f, a2 = 0.f;
  for (int n = threadIdx.x; n < NN; n += 256) {
    const float* p = x + ((size_t)b * NN + n) * 3;
    a0 += p[0]; a1 += p[1]; a2 += p[2];
  }
  sm[threadIdx.x][0] = a0; sm[threadIdx.x][1] = a1; sm[threadIdx.x][2] = a2;
  __syncthreads();
  for (int s = 128; s > 0; s >>= 1) {
    if (threadIdx.x < (unsigned)s) {
      sm[threadIdx.x][0] += sm[threadIdx.x + s][0];
      sm[threadIdx.x][1] += sm[threadIdx.x + s][1];
      sm[threadIdx.x][2] += sm[threadIdx.x + s][2];
    }
    __syncthreads();
  }
  if (threadIdx.x == 0) {
    center[b * 3 + 0] = sm[0][0] * (1.f / NN);
    center[b * 3 + 1] = sm[0][1] * (1.f / NN);
    center[b * 3 + 2] = sm[0][2] * (1.f / NN);
  }
}

// ---------------------------------------------------------------------------
// KNN: per query point keep K=20 smallest squared distances (self included,
// matching top_k on negd). idx holds within-batch neighbor indices.
// ---------------------------------------------------------------------------
__global__ __launch_bounds__(256) void knn_kernel(const float* __restrict__ x,
                                                  int* __restrict__ idx) {
  int b = blockIdx.x >> 4;                       // 16 blocks per batch
  int n = ((blockIdx.x & 15) << 8) + threadIdx.x;
  const float* xb = x + (size_t)b * NN * 3;
  float qx = xb[n * 3 + 0], qy = xb[n * 3 + 1], qz = xb[n * 3 + 2];
  float bd[NK];
  int   bi[NK];
#pragma unroll
  for (int t = 0; t < NK; ++t) { bd[t] = 3.0e38f; bi[t] = 0; }

  __shared__ float tile[256 * 3];
  for (int t0 = 0; t0 < NN; t0 += 256) {
    int j = t0 + threadIdx.x;
    tile[threadIdx.x * 3 + 0] = xb[j * 3 + 0];
    tile[threadIdx.x * 3 + 1] = xb[j * 3 + 1];
    tile[threadIdx.x * 3 + 2] = xb[j * 3 + 2];
    __syncthreads();
    for (int jj = 0; jj < 256; ++jj) {
      float dx = tile[jj * 3 + 0] - qx;
      float dy = tile[jj * 3 + 1] - qy;
      float dz = tile[jj * 3 + 2] - qz;
      float d2 = dx * dx + dy * dy + dz * dz;
      if (d2 < bd[NK - 1]) {
        bd[NK - 1] = d2; bi[NK - 1] = t0 + jj;
#pragma unroll
        for (int t = NK - 1; t > 0; --t) {
          if (bd[t] < bd[t - 1]) {
            float tf = bd[t]; bd[t] = bd[t - 1]; bd[t - 1] = tf;
            int   ti = bi[t]; bi[t] = bi[t - 1]; bi[t - 1] = ti;
          }
        }
      }
    }
    __syncthreads();
  }
  int* out = idx + ((size_t)b * NN + n) * NK;
#pragma unroll
  for (int t = 0; t < NK; ++t) out[t] = bi[t];
}

// ---------------------------------------------------------------------------
// Rotation-equivariant 5x3 edge feature
// ---------------------------------------------------------------------------
__device__ __forceinline__ void make_feat(const float* __restrict__ xb,
                                          const float* __restrict__ ctr,
                                          int n, int j, float f[5][3]) {
  float xi0 = xb[n * 3 + 0], xi1 = xb[n * 3 + 1], xi2 = xb[n * 3 + 2];
  float xj0 = xb[j * 3 + 0], xj1 = xb[j * 3 + 1], xj2 = xb[j * 3 + 2];
  float c0 = ctr[0], c1 = ctr[1], c2 = ctr[2];
  float d0 = xj0 - xi0, d1 = xj1 - xi1, d2 = xj2 - xi2;
  float a0 = xi0 - c0, a1 = xi1 - c1, a2 = xi2 - c2;   // xc
  float g0 = xj0 - c0, g1 = xj1 - c1, g2 = xj2 - c2;   // fc
  f[0][0] = d0; f[0][1] = d1; f[0][2] = d2;
  f[1][0] = a0; f[1][1] = a1; f[1][2] = a2;
  f[2][0] = d1 * a2 - d2 * a1; f[2][1] = d2 * a0 - d0 * a2; f[2][2] = d0 * a1 - d1 * a0;
  f[3][0] = g0; f[3][1] = g1; f[3][2] = g2;
  f[4][0] = g1 * a2 - g2 * a1; f[4][1] = g2 * a0 - g0 * a2; f[4][2] = g0 * a1 - g1 * a0;
}

__global__ void zero_stats_kernel(float* __restrict__ stats) {
  stats[threadIdx.x] = 0.f;   // 128 threads
}

// ---------------------------------------------------------------------------
// BN stats for the first (feature) layer: per-channel sum/sumsq of |Wf@feat|+eps
// over B*N*K samples. 256 threads = 4 samples x 64 channels.
// ---------------------------------------------------------------------------
__global__ __launch_bounds__(256) void feat_stats_kernel(
    const float* __restrict__ x, const int* __restrict__ idx,
    const float* __restrict__ center, const float* __restrict__ Wf,
    float* __restrict__ stats) {
  int c = threadIdx.x & 63;
  int s = threadIdx.x >> 6;
  long samp = (long)blockIdx.x * 4 + s;      // < B*N*K
  int k = (int)(samp % NK);
  long t = samp / NK;
  int n = (int)(t & (NN - 1));
  int b = (int)(t >> 12);
  const float* xb = x + (size_t)b * NN * 3;
  int j = idx[((size_t)b * NN + n) * NK + k];
  float f[5][3];
  make_feat(xb, center + b * 3, n, j, f);
  float p0 = 0.f, p1 = 0.f, p2 = 0.f;
#pragma unroll
  for (int c5 = 0; c5 < 5; ++c5) {
    float w = Wf[c * 5 + c5];
    p0 += w * f[c5][0]; p1 += w * f[c5][1]; p2 += w * f[c5][2];
  }
  float nv = sqrtf(p0 * p0 + p1 * p1 + p2 * p2) + EPSF;
  __shared__ float s1[4][64];
  __shared__ float s2[4][64];
  s1[s][c] = nv; s2[s][c] = nv * nv;
  __syncthreads();
  if (threadIdx.x < 64) {
    int cc = threadIdx.x;
    float a = s1[0][cc] + s1[1][cc] + s1[2][cc] + s1[3][cc];
    float q = s2[0][cc] + s2[1][cc] + s2[2][cc] + s2[3][cc];
    atomicAdd(&stats[cc], a);
    atomicAdd(&stats[64 + cc], q);
  }
}

__device__ __forceinline__ void vn_lrelu3(float& p0, float& p1, float& p2,
                                          float d0, float d1, float d2) {
  float dot = p0 * d0 + p1 * d1 + p2 * d2;
  if (dot < 0.f) {
    float dsq = d0 * d0 + d1 * d1 + d2 * d2;
    float fct = dot / (dsq + EPSF);
    p0 -= fct * d0; p1 -= fct * d1; p2 -= fct * d2;
  }
}

// ---------------------------------------------------------------------------
// First VN-lrelu (5->64) + mean over K  ->  x1_1 (B,64,3,N)
// one thread per (b,c,n), loop over K neighbors
// ---------------------------------------------------------------------------
__global__ __launch_bounds__(256) void feat_lrelu_kernel(
    const float* __restrict__ x, const int* __restrict__ idx,
    const float* __restrict__ center, const float* __restrict__ Wf,
    const float* __restrict__ Wd, const float* __restrict__ g,
    const float* __restrict__ bb, const float* __restrict__ stats,
    float* __restrict__ out) {
  int gid = blockIdx.x * 256 + threadIdx.x;       // B*NC*NN threads
  int n = gid & (NN - 1);
  int c = (gid >> 12) & 63;
  int b = gid >> 18;
  const float cnt = (float)NB * NN * NK;
  float mean = stats[c] / cnt;
  float var  = stats[64 + c] / cnt - mean * mean;
  float inv  = rsqrtf(var + 1e-5f);
  float gc = g[c], bc = bb[c];
  float wf[5], wd[5];
#pragma unroll
  for (int c5 = 0; c5 < 5; ++c5) { wf[c5] = Wf[c * 5 + c5]; wd[c5] = Wd[c * 5 + c5]; }
  const float* xb = x + (size_t)b * NN * 3;
  const int* id = idx + ((size_t)b * NN + n) * NK;
  float acc0 = 0.f, acc1 = 0.f, acc2 = 0.f;
  for (int k = 0; k < NK; ++k) {
    float f[5][3];
    make_feat(xb, center + b * 3, n, id[k], f);
    float p0 = 0.f, p1 = 0.f, p2 = 0.f, d0 = 0.f, d1 = 0.f, d2 = 0.f;
#pragma unroll
    for (int c5 = 0; c5 < 5; ++c5) {
      p0 += wf[c5] * f[c5][0]; p1 += wf[c5] * f[c5][1]; p2 += wf[c5] * f[c5][2];
      d0 += wd[c5] * f[c5][0]; d1 += wd[c5] * f[c5][1]; d2 += wd[c5] * f[c5][2];
    }
    float nv = sqrtf(p0 * p0 + p1 * p1 + p2 * p2) + EPSF;
    float nbn = (nv - mean) * inv * gc + bc;
    float scl = nbn / nv;
    p0 *= scl; p1 *= scl; p2 *= scl;
    vn_lrelu3(p0, p1, p2, d0, d1, d2);
    acc0 += p0; acc1 += p1; acc2 += p2;
  }
  size_t base = ((size_t)(b * NC + c)) * COLSB + n;
  out[base]          = acc0 * (1.f / NK);
  out[base + NN]     = acc1 * (1.f / NK);
  out[base + 2 * NN] = acc2 * (1.f / NK);
}

// ---------------------------------------------------------------------------
// WMMA GEMM: Y_i(64 x 12288, per batch) = W_i(64x64) @ X(64 x 12288, per batch)
// for i < NMAT (compile-time; up to 3 weight matrices share the B fragments).
// One wave32 computes a 16(rows)x32(cols) tile: per k-step 2*NMAT WMMAs
// (V_WMMA_F32_16X16X4_F32) against 4 in-loop loads; all A fragments are
// preloaded into registers before the K-loop.
// A layout (16x4 f32): lane = M (mod 16); VGPR0/1 = K{0,1} (lanes 0-15),
// K{2,3} (lanes 16-31). B layout mirrors with lane = N. D per ISA 16x16 f32.
// ---------------------------------------------------------------------------
template <int NMAT>
__global__ __launch_bounds__(128) void gemm64_wmma_kernel(
    const float* __restrict__ W0, const float* __restrict__ W1,
    const float* __restrict__ W2, const float* __restrict__ X,
    float* __restrict__ Y0, float* __restrict__ Y1, float* __restrict__ Y2) {
  int wave = ((blockIdx.x * 128 + threadIdx.x) >> 5);   // 0..6143
  int lane = threadIdx.x & 31;
  int rowTile = wave & 3;            // 4 row tiles (64 rows)
  int colPair = wave >> 2;           // 1536 tiles of 32 cols (B*12288/32)
  int m0 = colPair * 32;
  int b  = m0 / COLSB;               // 32-col tile never crosses a batch
  int r0 = m0 - b * COLSB;
  const float* Xb = X + (size_t)b * NC * COLSB;
  int o0   = rowTile * 16;
  int half = lane >> 4;
  int l16  = lane & 15;
  int row  = o0 + l16;

  const float* Ws[3] = {W0, W1, W2};
  float* Ys[3] = {Y0, Y1, Y2};

  // Preload A fragments (weights) for all 16 k-steps, all matrices.
  v2f af[NMAT][16];
#pragma unroll
  for (int m = 0; m < NMAT; ++m) {
#pragma unroll
    for (int ks = 0; ks < 16; ++ks) {
      int kk = ks * 4 + half * 2;
      af[m][ks].x = Ws[m][row * 64 + kk];
      af[m][ks].y = Ws[m][row * 64 + kk + 1];
    }
  }

  v8f acc[NMAT][2];
#pragma unroll
  for (int m = 0; m < NMAT; ++m) { acc[m][0] = (v8f){}; acc[m][1] = (v8f){}; }

#pragma unroll
  for (int ks = 0; ks < 16; ++ks) {
    int kk = ks * 4 + half * 2;
    v2f bf0, bf1;
    bf0.x = Xb[(size_t)kk * COLSB + r0 + l16];
    bf0.y = Xb[(size_t)(kk + 1) * COLSB + r0 + l16];
    bf1.x = Xb[(size_t)kk * COLSB + r0 + 16 + l16];
    bf1.y = Xb[(size_t)(kk + 1) * COLSB + r0 + 16 + l16];
#pragma unroll
    for (int m = 0; m < NMAT; ++m) {
      acc[m][0] = __builtin_amdgcn_wmma_f32_16x16x4_f32(false, af[m][ks], false, bf0,
                                                        (short)0, acc[m][0], false, false);
      acc[m][1] = __builtin_amdgcn_wmma_f32_16x16x4_f32(false, af[m][ks], false, bf1,
                                                        (short)0, acc[m][1], false, false);
    }
  }

  size_t base = (size_t)b * NC * COLSB + r0 + l16;
#pragma unroll
  for (int m = 0; m < NMAT; ++m) {
#pragma unroll
    for (int v = 0; v < 8; ++v) {
      int orow = o0 + v + half * 8;
      Ys[m][base + (size_t)orow * COLSB]      = acc[m][0][v];
      Ys[m][base + (size_t)orow * COLSB + 16] = acc[m][1][v];
    }
  }
}

// ---------------------------------------------------------------------------
// BN stats for 64-channel layers: sum/sumsq of |p|+eps over (B,N) per channel.
// ---------------------------------------------------------------------------
__global__ __launch_bounds__(256) void stats64_kernel(const float* __restrict__ P,
                                                      float* __restrict__ stats) {
  int c = threadIdx.x & 63;
  int s = threadIdx.x >> 6;
  int samp = blockIdx.x * 4 + s;         // < B*N
  int n = samp & (NN - 1);
  int b = samp >> 12;
  size_t base = ((size_t)(b * NC + c)) * COLSB + n;
  float p0 = P[base], p1 = P[base + NN], p2 = P[base + 2 * NN];
  float nv = sqrtf(p0 * p0 + p1 * p1 + p2 * p2) + EPSF;
  __shared__ float s1[4][64];
  __shared__ float s2[4][64];
  s1[s][c] = nv; s2[s][c] = nv * nv;
  __syncthreads();
  if (threadIdx.x < 64) {
    int cc = threadIdx.x;
    float a = s1[0][cc] + s1[1][cc] + s1[2][cc] + s1[3][cc];
    float q = s2[0][cc] + s2[1][cc] + s2[2][cc] + s2[3][cc];
    atomicAdd(&stats[cc], a);
    atomicAdd(&stats[64 + cc], q);
  }
}

// ---------------------------------------------------------------------------
// Apply BN + VN-lrelu + residual:  OUT = vn_lrelu(BN(P), D) + RES
// ---------------------------------------------------------------------------
__global__ __launch_bounds__(256) void apply_lrelu_kernel(
    const float* __restrict__ P, const float* __restrict__ D,
    const float* __restrict__ stats, const float* __restrict__ g,
    const float* __restrict__ bb, const float* __restrict__ RES,
    float* __restrict__ OUT) {
  int gid = blockIdx.x * 256 + threadIdx.x;      // B*NC*NN
  int n = gid & (NN - 1);
  int c = (gid >> 12) & 63;
  int b = gid >> 18;
  const float cnt = (float)NB * NN;
  float mean = stats[c] / cnt;
  float var  = stats[64 + c] / cnt - mean * mean;
  float inv  = rsqrtf(var + 1e-5f);
  size_t base = ((size_t)(b * NC + c)) * COLSB + n;
  float p0 = P[base], p1 = P[base + NN], p2 = P[base + 2 * NN];
  float nv = sqrtf(p0 * p0 + p1 * p1 + p2 * p2) + EPSF;
  float nbn = (nv - mean) * inv * g[c] + bb[c];
  float scl = nbn / nv;
  p0 *= scl; p1 *= scl; p2 *= scl;
  float d0 = D[base], d1 = D[base + NN], d2 = D[base + 2 * NN];
  vn_lrelu3(p0, p1, p2, d0, d1, d2);
  OUT[base]          = p0 + RES[base];
  OUT[base + NN]     = p1 + RES[base + NN];
  OUT[base + 2 * NN] = p2 + RES[base + 2 * NN];
}

__global__ __launch_bounds__(256) void add_kernel(const float* __restrict__ A,
                                                  const float* __restrict__ B,
                                                  float* __restrict__ O) {
  size_t i = (size_t)blockIdx.x * 256 + threadIdx.x;
  O[i] = A[i] + B[i];
}

// ---------------------------------------------------------------------------
// Per-point channel attention: s=QK^T/sqrt(3) over 64 channels (inner dim 3),
// softmax, A@V; outputs channel-mean of (attn + Xres) -> (B,3,N).
// One block (64 threads = 2 waves) per point.
// ---------------------------------------------------------------------------
__global__ __launch_bounds__(64) void attn_kernel(
    const float* __restrict__ Q, const float* __restrict__ K,
    const float* __restrict__ V, const float* __restrict__ Xres,
    float* __restrict__ outmean) {
  int i = threadIdx.x;                // channel
  int n = blockIdx.x & (NN - 1);
  int b = blockIdx.x >> 12;
  size_t base = ((size_t)(b * NC + i)) * COLSB + n;
  float qi0 = Q[base], qi1 = Q[base + NN], qi2 = Q[base + 2 * NN];
  __shared__ float ks[64][3];
  __shared__ float vs[64][3];
  ks[i][0] = K[base]; ks[i][1] = K[base + NN]; ks[i][2] = K[base + 2 * NN];
  vs[i][0] = V[base]; vs[i][1] = V[base + NN]; vs[i][2] = V[base + 2 * NN];
  __syncthreads();
  const float SCL = 0.5773502691896258f;   // 1/sqrt(3)
  float mx = -3.0e38f;
  for (int j = 0; j < 64; ++j) {
    float s = (qi0 * ks[j][0] + qi1 * ks[j][1] + qi2 * ks[j][2]) * SCL;
    mx = fmaxf(mx, s);
  }
  float se = 0.f, o0 = 0.f, o1 = 0.f, o2 = 0.f;
  for (int j = 0; j < 64; ++j) {
    float s = (qi0 * ks[j][0] + qi1 * ks[j][1] + qi2 * ks[j][2]) * SCL;
    float e = expf(s - mx);
    se += e;
    o0 += e * vs[j][0]; o1 += e * vs[j][1]; o2 += e * vs[j][2];
  }
  float rse = 1.f / se;
  float res[3];
  res[0] = o0 * rse + Xres[base];
  res[1] = o1 * rse + Xres[base + NN];
  res[2] = o2 * rse + Xres[base + 2 * NN];
  __shared__ float red[64];
  for (int d = 0; d < 3; ++d) {
    red[i] = res[d];
    __syncthreads();
    if (i == 0) {
      float t = 0.f;
      for (int j = 0; j < 64; ++j) t += red[j];
      outmean[((size_t)b * 3 + d) * NN + n] = t * (1.f / 64.f);
    }
    __syncthreads();
  }
}

// ---------------------------------------------------------------------------
// Final features: ojld (gathered distance max), cos (gathered dot max), nrm.
// ---------------------------------------------------------------------------
__global__ __launch_bounds__(256) void final_kernel(
    const float* __restrict__ x12, const float* __restrict__ x22,
    const float* __restrict__ x13, const float* __restrict__ x23,
    const float* __restrict__ x14, const float* __restrict__ x24,
    const float* __restrict__ m51, const float* __restrict__ m52,
    const int* __restrict__ idx, float* __restrict__ out) {
  int gid = blockIdx.x * 256 + threadIdx.x;     // B*NC*NN
  int n = gid & (NN - 1);
  int c = (gid >> 12) & 63;
  int b = gid >> 18;
  size_t base = ((size_t)(b * NC + c)) * COLSB + n;
  float s2v[3], s4v[3];
#pragma unroll
  for (int d = 0; d < 3; ++d) {
    s2v[d] = x12[base + (size_t)d * NN] + x22[base + (size_t)d * NN];
    s4v[d] = x14[base + (size_t)d * NN] + x24[base + (size_t)d * NN];
  }
  float nrm = sqrtf(s4v[0] * s4v[0] + s4v[1] * s4v[1] + s4v[2] * s4v[2]);
  const int* id = idx + ((size_t)b * NN + n) * NK;
  float oj = -3.0e38f, cs = -3.0e38f;
  for (int k = 0; k < NK; ++k) {
    int j = id[k];
    size_t gb = ((size_t)(b * NC + c)) * COLSB + j;
    float dd = 0.f;
#pragma unroll
    for (int d = 0; d < 3; ++d) {
      float diff = s2v[d] - (x13[gb + (size_t)d * NN] + x23[gb + (size_t)d * NN]) + EPSF;
      dd += diff * diff;
    }
    oj = fmaxf(oj, sqrtf(dd));
    float dotv = 0.f;
#pragma unroll
    for (int d = 0; d < 3; ++d) {
      float x5 = m51[((size_t)b * 3 + d) * NN + j] + m52[((size_t)b * 3 + d) * NN + j];
      dotv += s4v[d] * x5;
    }
    cs = fmaxf(cs, dotv);
  }
  size_t ob = ((size_t)b * NN + n) * 192;
  out[ob + c]        = oj;
  out[ob + 64 + c]   = cs;
  out[ob + 128 + c]  = nrm;
}

// ---------------------------------------------------------------------------
// Host-side pipeline
// ---------------------------------------------------------------------------
extern "C" void kernel_launch(void* const* d_in, const int* in_sizes, int n_in,
                              void* d_out, int out_size, void* d_ws, size_t ws_size,
                              hipStream_t stream) {
  (void)in_sizes; (void)n_in; (void)out_size; (void)ws_size;
  const float* x     = (const float*)d_in[0];
  const float* Wp1_f = (const float*)d_in[1];
  const float* Wp1_d = (const float*)d_in[2];
  const float* gp1   = (const float*)d_in[3];
  const float* bp1   = (const float*)d_in[4];
  const float* W11_f = (const float*)d_in[5];
  const float* W11_d = (const float*)d_in[6];
  const float* W12_f = (const float*)d_in[7];
  const float* W12_d = (const float*)d_in[8];
  const float* Wq1   = (const float*)d_in[9];
  const float* Wk1   = (const float*)d_in[10];
  const float* Wv1   = (const float*)d_in[11];
  const float* W21_f = (const float*)d_in[12];
  const float* W21_d = (const float*)d_in[13];
  const float* Wq2   = (const float*)d_in[14];
  const float* Wk2   = (const float*)d_in[15];
  const float* Wv2   = (const float*)d_in[16];
  const float* g11   = (const float*)d_in[17];
  const float* g12   = (const float*)d_in[18];
  const float* g21   = (const float*)d_in[19];
  const float* b11   = (const float*)d_in[20];
  const float* b12   = (const float*)d_in[21];
  const float* b21   = (const float*)d_in[22];

  float* wsf = (float*)d_ws;
  int*   idx = (int*)d_ws;                      // B*N*K ints
  size_t o = (size_t)NB * NN * NK;              // 327,680
  float* center = wsf + o; o += 16;
  float* stats  = wsf + o; o += 128;
  float* m51 = wsf + o; o += (size_t)NB * 3 * NN;
  float* m52 = wsf + o; o += (size_t)NB * 3 * NN;
  o = (o + 255) & ~(size_t)255;
  float* t11 = wsf + o; o += TSZ;
  float* t12 = wsf + o; o += TSZ;
  float* t13 = wsf + o; o += TSZ;
  float* t14 = wsf + o; o += TSZ;
  float* t22 = wsf + o; o += TSZ;
  float* t23 = wsf + o; o += TSZ;
  float* t24 = wsf + o; o += TSZ;
  float* s0  = wsf + o; o += TSZ;
  float* s1  = wsf + o; o += TSZ;
  float* s2  = wsf + o; o += TSZ;

  const int GEMM_BLK = 1536;   // 6144 waves (4 row x 1536 colpair) / 4 per block
  const int PT_BLK   = 4096;   // B*NC*NN / 256
  const int ST_BLK   = 4096;   // B*N / 4

  // KNN + geometry
  center_kernel<<<NB, 256, 0, stream>>>(x, center);
  knn_kernel<<<64, 256, 0, stream>>>(x, idx);

  // x1 = vn_lrelu(feat); x1_1 = mean over K
  zero_stats_kernel<<<1, 128, 0, stream>>>(stats);
  feat_stats_kernel<<<(NB * NN * NK) / 4, 256, 0, stream>>>(x, idx, center, Wp1_f, stats);
  feat_lrelu_kernel<<<PT_BLK, 256, 0, stream>>>(x, idx, center, Wp1_f, Wp1_d, gp1, bp1,
                                                stats, t11);
  // x1_2 = lrelu(x1_1; W11) + x1_1
  gemm64_wmma_kernel<2><<<GEMM_BLK, 128, 0, stream>>>(W11_f, W11_d, W11_d, t11, s0, s1, s1);
  zero_stats_kernel<<<1, 128, 0, stream>>>(stats);
  stats64_kernel<<<ST_BLK, 256, 0, stream>>>(s0, stats);
  apply_lrelu_kernel<<<PT_BLK, 256, 0, stream>>>(s0, s1, stats, g11, b11, t11, t12);

  // x1_3 = lrelu(x1_2; W12) + x1_2
  gemm64_wmma_kernel<2><<<GEMM_BLK, 128, 0, stream>>>(W12_f, W12_d, W12_d, t12, s0, s1, s1);
  zero_stats_kernel<<<1, 128, 0, stream>>>(stats);
  stats64_kernel<<<ST_BLK, 256, 0, stream>>>(s0, stats);
  apply_lrelu_kernel<<<PT_BLK, 256, 0, stream>>>(s0, s1, stats, g12, b12, t12, t13);

  // x1_4 = x1_2 + x1_3
  add_kernel<<<(int)(TSZ / 256), 256, 0, stream>>>(t12, t13, t14);

  // x1_5 = mean_c(attn(x1_4) + x1_4)
  gemm64_wmma_kernel<3><<<GEMM_BLK, 128, 0, stream>>>(Wq1, Wk1, Wv1, t14, s0, s1, s2);
  attn_kernel<<<NB * NN, 64, 0, stream>>>(s0, s1, s2, t14, m51);

  // x2_2 = lrelu(x1_4; W21) + x1_4
  gemm64_wmma_kernel<2><<<GEMM_BLK, 128, 0, stream>>>(W21_f, W21_d, W21_d, t14, s0, s1, s1);
  zero_stats_kernel<<<1, 128, 0, stream>>>(stats);
  stats64_kernel<<<ST_BLK, 256, 0, stream>>>(s0, stats);
  apply_lrelu_kernel<<<PT_BLK, 256, 0, stream>>>(s0, s1, stats, g21, b21, t14, t22);

  // x2_3 = lrelu(x2_2; W21) + x2_2
  gemm64_wmma_kernel<2><<<GEMM_BLK, 128, 0, stream>>>(W21_f, W21_d, W21_d, t22, s0, s1, s1);
  zero_stats_kernel<<<1, 128, 0, stream>>>(stats);
  stats64_kernel<<<ST_BLK, 256, 0, stream>>>(s0, stats);
  apply_lrelu_kernel<<<PT_BLK, 256, 0, stream>>>(s0, s1, stats, g21, b21, t22, t23);

  // x2_4 = x2_2 + x2_3
  add_kernel<<<(int)(TSZ / 256), 256, 0, stream>>>(t22, t23, t24);

  // x2_5 = mean_c(attn(x2_4) + x2_4)
  gemm64_wmma_kernel<3><<<GEMM_BLK, 128, 0, stream>>>(Wq2, Wk2, Wv2, t24, s0, s1, s2);
  attn_kernel<<<NB * NN, 64, 0, stream>>>(s0, s1, s2, t24, m52);

  // final gather features -> (B,N,192)
  final_kernel<<<PT_BLK, 256, 0, stream>>>(t12, t22, t13, t23, t14, t24, m51, m52,
                                           idx, (float*)d_out);
}